// RefWrapperModule_7232724927035
// MI455X (gfx1250) — compile-verified
//
#include <hip/hip_runtime.h>
#include <stdint.h>

typedef unsigned int v4u __attribute__((ext_vector_type(4)));
typedef int          v4i __attribute__((ext_vector_type(4)));
typedef int          v8i __attribute__((ext_vector_type(8)));

#define NDIM    512
#define NCH     16
#define NNZ_TOT 4096
#define TPB     256
#define ROW_F   (NDIM * NCH)        // 8192 fp32 per row per tensor (32 KiB)
#define ROW_V4  (ROW_F / 4)
#define V4PT    (ROW_V4 / TPB)      // 8 float4 per thread

// LDS offsets are workgroup-relative; low 32 bits of a generic pointer to a
// __shared__ object are exactly that offset (aperture lives in the high bits).
__device__ __forceinline__ uint32_t lds_off(const void* p) {
    return (uint32_t)(uintptr_t)p;
}

// ---- Tensor Data Mover descriptor for one contiguous 32 KiB fp32 row ------
// D# group 0 (ISA §8.3): count=1 | lds_addr | global_addr[56:0] | type=2
__device__ __forceinline__ v4u tdm_g0(uint32_t lds_addr, uint64_t gaddr) {
    v4u g;
    g.x = 0x1u;                                   // count=1, user descriptor
    g.y = lds_addr;                               // LDS byte address
    g.z = (uint32_t)gaddr;                        // global_addr[31:0]
    g.w = (uint32_t)(gaddr >> 32) | (2u << 30);   // global_addr[56:32] | type=2
    return g;
}
// D# group 1 (ISA §8.4): mask=0, data_size=4B, tensor_dim0=tile_dim0=ROW_F,
// tensor_dim1=1, tile_dim1=tile_dim2=1, stride0=ROW_F, stride1=0.
__device__ __forceinline__ v8i tdm_g1() {
    v8i g;
    g[0] = (int)(2u << 16);                            // data_size=4B
    g[1] = (int)((ROW_F & 0xFFFFu) << 16);             // tensor_dim0[15:0] @63:48
    g[2] = (int)((ROW_F >> 16) | (1u << 16));          // tdim0[31:16] | tdim1[15:0]=1
    g[3] = (int)((ROW_F & 0xFFFFu) << 16);             // tdim1[31:16]=0 | tile_dim0
    g[4] = (int)(1u | (1u << 16));                     // tile_dim1=1 | tile_dim2=1
    g[5] = (int)ROW_F;                                 // tensor_dim0_stride[31:0]
    g[6] = 0;                                          // stride0[47:32] | stride1[15:0]
    g[7] = 0;                                          // stride1[47:16]
    return g;
}
// D# groups 2/3 (ISA §8.5/8.6): trailing dims collapsed to 1, strides 0.
__device__ __forceinline__ v4i tdm_g2() {
    v4i g; g[0] = 1; g[1] = 1; g[2] = 0; g[3] = (int)(1u << 16); return g;
}
__device__ __forceinline__ v4i tdm_g3() {
    v4i g; g[0] = 0; g[1] = (int)(1u << 16); g[2] = (int)(1u << 16); g[3] = 0; return g;
}

__device__ __forceinline__ void tdm_row_load(uint32_t lds_addr, const void* gaddr) {
    v4u g0 = tdm_g0(lds_addr, (uint64_t)(uintptr_t)gaddr);
    v8i g1 = tdm_g1();
    v4i g2 = tdm_g2(), g3 = tdm_g3();
#if __has_include(<hip/amd_detail/amd_gfx1250_TDM.h>)
    v8i zz = {0, 0, 0, 0, 0, 0, 0, 0};
    __builtin_amdgcn_tensor_load_to_lds(g0, g1, g2, g3, zz, 0);
#else
    __builtin_amdgcn_tensor_load_to_lds(g0, g1, g2, g3, 0);
#endif
}
__device__ __forceinline__ void tdm_row_store(uint32_t lds_addr, void* gaddr) {
    v4u g0 = tdm_g0(lds_addr, (uint64_t)(uintptr_t)gaddr);
    v8i g1 = tdm_g1();
    v4i g2 = tdm_g2(), g3 = tdm_g3();
#if __has_include(<hip/amd_detail/amd_gfx1250_TDM.h>)
    v8i zz = {0, 0, 0, 0, 0, 0, 0, 0};
    __builtin_amdgcn_tensor_store_from_lds(g0, g1, g2, g3, zz, 0);
#else
    __builtin_amdgcn_tensor_store_from_lds(g0, g1, g2, g3, 0);
#endif
}

// ---- prep: pack (i1*NCH, i2*NCH, io*NCH, scale) into one int4 per path ----
__global__ void pack_paths(const float* __restrict__ scale,
                           const int* __restrict__ i1,
                           const int* __restrict__ i2,
                           const int* __restrict__ io,
                           int4* __restrict__ ews) {
    const int k = blockIdx.x * blockDim.x + threadIdx.x;
    if (k < NNZ_TOT) {
        int4 e;
        e.x = i1[k] * NCH;
        e.y = i2[k] * NCH;
        e.z = io[k] * NCH;
        e.w = __float_as_int(scale[k]);
        ews[k] = e;
    }
}

template <bool PACKED>
__global__ __launch_bounds__(TPB, 1)
void sparse_tp_kernel(const float* __restrict__ x,
                      const float* __restrict__ y,
                      float*       __restrict__ out,
                      const int4*  __restrict__ ews,
                      const float* __restrict__ scale,
                      const int*   __restrict__ idx1,
                      const int*   __restrict__ idx2,
                      const int*   __restrict__ idxo) {
    __shared__ float xs[ROW_F];   // 32 KiB x-tile
    __shared__ float ys[ROW_F];   // 32 KiB y-tile
    __shared__ float os[ROW_F];   // 32 KiB accumulation tile

    const int n   = blockIdx.x;
    const int tid = threadIdx.x;

    // Phase 1: TDM DMA of both input tiles (one descriptor each, issued by
    // two different waves so the DMAs run unordered/parallel), while all
    // threads zero the accumulation tile.
    if (tid == 0)  tdm_row_load(lds_off(xs), x + (size_t)n * ROW_F);
    if (tid == 32) tdm_row_load(lds_off(ys), y + (size_t)n * ROW_F);

    float4* os4 = (float4*)os;
#pragma unroll
    for (int i = 0; i < V4PT; ++i)
        os4[tid + i * TPB] = make_float4(0.f, 0.f, 0.f, 0.f);

    __builtin_amdgcn_s_wait_tensorcnt(0);   // no-op for non-issuing waves
    __syncthreads();

    // Phase 2: walk a contiguous slice of the sorted nnz list.
    // thread = (kslot, c): 16 lanes share each k (one b128 broadcast load),
    // sorted index_out => register-accumulated runs, ds_add_f32 at boundaries.
    const int c     = tid & (NCH - 1);
    const int kslot = tid >> 4;
    const int chunk = NNZ_TOT / (TPB / NCH);   // 256
    const int k0    = kslot * chunk;
    const int k1    = k0 + chunk;

    int   cur = -1;
    float acc = 0.f;
    if (PACKED) {
        for (int k = k0; k < k1; ++k) {
            const int4 e = ews[k];
            if (e.z != cur) {
                if (cur >= 0) atomicAdd(&os[cur + c], acc);
                cur = e.z;
                acc = 0.f;
            }
            acc += __int_as_float(e.w) * xs[e.x + c] * ys[e.y + c];
        }
        if (cur >= 0) atomicAdd(&os[cur + c], acc);
    } else {
        for (int k = k0; k < k1; ++k) {
            const int o = idxo[k] * NCH;
            if (o != cur) {
                if (cur >= 0) atomicAdd(&os[cur + c], acc);
                cur = o;
                acc = 0.f;
            }
            acc += scale[k] * xs[idx1[k] * NCH + c] * ys[idx2[k] * NCH + c];
        }
        if (cur >= 0) atomicAdd(&os[cur + c], acc);
    }

    __syncthreads();   // all LDS atomics drained (DScnt) before TDM reads LDS

    // Phase 3: TDM DMA of the finished tile back to HBM.
    if (tid == 0) tdm_row_store(lds_off(os), out + (size_t)n * ROW_F);
    __builtin_amdgcn_s_wait_tensorcnt(0);
}

extern "C" void kernel_launch(void* const* d_in, const int* in_sizes, int n_in,
                              void* d_out, int out_size, void* d_ws, size_t ws_size,
                              hipStream_t stream) {
    const float* x     = (const float*)d_in[0];
    const float* y     = (const float*)d_in[1];
    const float* scale = (const float*)d_in[2];
    const int*   idx1  = (const int*)d_in[3];
    const int*   idx2  = (const int*)d_in[4];
    const int*   idxo  = (const int*)d_in[5];
    float* out = (float*)d_out;

    const int n = in_sizes[0] / ROW_F;   // 2048 batch rows, one block each

    if (ws_size >= NNZ_TOT * sizeof(int4)) {
        int4* ews = (int4*)d_ws;
        pack_paths<<<NNZ_TOT / TPB, TPB, 0, stream>>>(scale, idx1, idx2, idxo, ews);
        sparse_tp_kernel<true><<<n, TPB, 0, stream>>>(x, y, out, ews,
                                                      scale, idx1, idx2, idxo);
    } else {
        sparse_tp_kernel<false><<<n, TPB, 0, stream>>>(x, y, out, nullptr,
                                                       scale, idx1, idx2, idxo);
    }
}